// LoLAState_52493090292199
// MI455X (gfx1250) — compile-verified
//
#include <hip/hip_runtime.h>
#include <hip/hip_bf16.h>

typedef float v2f __attribute__((ext_vector_type(2)));
typedef float v8f __attribute__((ext_vector_type(8)));

#define Bc 2
#define Hc 32
#define Gc 2048
#define Cc 2048
#define Nc 4096
#define Dd 128
#define Ff 128
#define BH (Bc * Hc)
#define CHUNK 32

// -------------------------------------------------------------------------
// Kernel 1: stable descending rank of the 4096 concatenated scores per head.
// rank[i] = #{ j : s[j] > s[i] || (s[j] == s[i] && j < i) }  (== stable argsort pos)
// Writes heap_score_new directly, plus compacted top/bottom source indices.
// Grid: (16, BH), block: 256. LDS: 16 KB of scores.
// -------------------------------------------------------------------------
__global__ __launch_bounds__(256) void rank_kernel(
    const float* __restrict__ heap_score, const float* __restrict__ win_score,
    float* __restrict__ out_score, int* __restrict__ top_idx, int* __restrict__ bot_idx) {
  __shared__ float sc[Nc];
  const int bh  = blockIdx.y;
  const int tid = threadIdx.x;
  const float* hs = heap_score + (size_t)bh * Gc;
  const float* ws = win_score  + (size_t)bh * Cc;
  for (int i = tid; i < Nc; i += 256)
    sc[i] = (i < Gc) ? hs[i] : ws[i - Gc];
  __syncthreads();

  const int i = blockIdx.x * 256 + tid;   // element this thread ranks
  const float s = sc[i];
  int rank = 0;
#pragma unroll 4
  for (int j = 0; j < Nc; j++) {
    const float t = sc[j];
    rank += (int)((t > s) || (t == s && j < i));
  }
  if (rank < Gc) {
    out_score[(size_t)bh * Gc + rank] = s;
    top_idx [(size_t)bh * Gc + rank] = i;
  } else {
    bot_idx [(size_t)bh * Cc + (rank - Gc)] = i;
  }
}

// -------------------------------------------------------------------------
// Kernel 2: gather K/V/FK rows of the top-2048 entries (bandwidth dominant).
// Grid: (Gc, BH), block: 128 (one thread per column). Fully coalesced.
// -------------------------------------------------------------------------
__global__ __launch_bounds__(128) void gather_kernel(
    const float* __restrict__ K_top, const float* __restrict__ V_top,
    const float* __restrict__ FK_top,
    const float* __restrict__ K_win, const float* __restrict__ V_win,
    const float* __restrict__ FK_win,
    const int* __restrict__ top_idx,
    float* __restrict__ K_out, float* __restrict__ V_out, float* __restrict__ FK_out) {
  const int g  = blockIdx.x;
  const int bh = blockIdx.y;
  const int t  = threadIdx.x;
  const int src = top_idx[(size_t)bh * Gc + g];
  const size_t dst = ((size_t)bh * Gc + g) * 128 + t;
  if (src < Gc) {
    const size_t so = ((size_t)bh * Gc + src) * 128 + t;
    K_out[dst]  = K_top[so];
    V_out[dst]  = V_top[so];
    FK_out[dst] = FK_top[so];
  } else {
    const size_t so = ((size_t)bh * Cc + (src - Gc)) * 128 + t;
    K_out[dst]  = K_win[so];
    V_out[dst]  = V_win[so];
    FK_out[dst] = FK_win[so];
  }
}

// -------------------------------------------------------------------------
// Kernel 3: H_sum[f,d] = sum_g FK[g,f] * V[g,d] over bottom rows (K=2048),
// plus S_sum[f] = sum_g FK[g,f]. One block (8 waves) per (b,h) head.
// Bottom rows staged through LDS in 32-row chunks; each wave owns one 16-row
// f-tile strip and accumulates 8 d-tiles with V_WMMA_F32_16X16X4_F32.
// A-fragment layout (ISA 7.12.2, fp32 16x4): lanes 0-15 -> K = k4+0/k4+1,
// lanes 16-31 -> K = k4+2/k4+3; B symmetric with N across lanes.
// -------------------------------------------------------------------------
__global__ __launch_bounds__(256) void hsum_kernel(
    const float* __restrict__ FK_top, const float* __restrict__ FK_win,
    const float* __restrict__ V_top,  const float* __restrict__ V_win,
    const int* __restrict__ bot_idx,
    float* __restrict__ Hout, float* __restrict__ Sout) {
  __shared__ float fk[CHUNK][Ff];   // 16 KB
  __shared__ float vv[CHUNK][Dd];   // 16 KB
  __shared__ int   srcs[CHUNK];

  const int bh   = blockIdx.x;
  const int tid  = threadIdx.x;
  const int wave = tid >> 5;        // 0..7 -> f-tile row
  const int lane = tid & 31;

  const float* FKt = FK_top + (size_t)bh * Gc * Ff;
  const float* FKw = FK_win + (size_t)bh * Cc * Ff;
  const float* Vt  = V_top  + (size_t)bh * Gc * Dd;
  const float* Vw  = V_win  + (size_t)bh * Cc * Dd;
  const int*   bid = bot_idx + (size_t)bh * Cc;

  v8f acc[8] = {};                  // 8 d-tiles of 16x16 f32
  float ssum = 0.0f;

  const int fbase = wave * 16;
  const int mlane = lane & 15;
  const int khalf = (lane >> 4) << 1;   // 0 for lanes 0-15, 2 for lanes 16-31

  for (int kb = 0; kb < Cc; kb += CHUNK) {
    if (tid < CHUNK) srcs[tid] = bid[kb + tid];
    __syncthreads();

    // Gather CHUNK rows of FK and V into LDS (coalesced 512B row segments).
    for (int idx = tid; idx < CHUNK * 128; idx += 256) {
      const int r = idx >> 7;
      const int c = idx & 127;
      const int s = srcs[r];
      if (s < Gc) {
        fk[r][c] = FKt[(size_t)s * Ff + c];
        vv[r][c] = Vt [(size_t)s * Dd + c];
      } else {
        fk[r][c] = FKw[(size_t)(s - Gc) * Ff + c];
        vv[r][c] = Vw [(size_t)(s - Gc) * Dd + c];
      }
    }
    __syncthreads();

    // S_sum partial: column sums of the FK chunk (waves 0-3, no lane divergence).
    if (tid < Ff) {
#pragma unroll 8
      for (int r = 0; r < CHUNK; r++) ssum += fk[r][tid];
    }

    // WMMA accumulation: K in steps of 4, wave-uniform control flow (EXEC all 1s).
    for (int k4 = 0; k4 < CHUNK; k4 += 4) {
      v2f a;
      a.x = fk[k4 + khalf + 0][fbase + mlane];
      a.y = fk[k4 + khalf + 1][fbase + mlane];
#pragma unroll
      for (int j = 0; j < 8; j++) {
        v2f b;
        b.x = vv[k4 + khalf + 0][j * 16 + mlane];
        b.y = vv[k4 + khalf + 1][j * 16 + mlane];
        acc[j] = __builtin_amdgcn_wmma_f32_16x16x4_f32(
            /*neg_a=*/false, a, /*neg_b=*/false, b,
            /*c_mod=*/(short)0, acc[j], /*reuse_a=*/false, /*reuse_b=*/false);
      }
    }
    __syncthreads();
  }

  // Store D-matrix: VGPR r, lanes 0-15 -> (M=r, N=lane); lanes 16-31 -> (M=8+r, N=lane-16).
  float* Hb = Hout + (size_t)bh * Ff * Dd;
  const int n    = lane & 15;
  const int moff = (lane >> 4) * 8;
#pragma unroll
  for (int j = 0; j < 8; j++) {
#pragma unroll
    for (int r = 0; r < 8; r++) {
      Hb[(size_t)(fbase + moff + r) * Dd + j * 16 + n] = acc[j][r];
    }
  }
  if (tid < Ff) Sout[(size_t)bh * Ff + tid] = ssum;
}

// -------------------------------------------------------------------------
// Host-side launcher.
// d_in order: K_top, V_top, FK_top, heap_score, K_win, V_win, FK_win, win_score
// d_out: heap_score_new | K_top_new | V_top_new | FK_top_new | H_sum | S_sum
// -------------------------------------------------------------------------
extern "C" void kernel_launch(void* const* d_in, const int* in_sizes, int n_in,
                              void* d_out, int out_size, void* d_ws, size_t ws_size,
                              hipStream_t stream) {
  const float* K_top      = (const float*)d_in[0];
  const float* V_top      = (const float*)d_in[1];
  const float* FK_top     = (const float*)d_in[2];
  const float* heap_score = (const float*)d_in[3];
  const float* K_win      = (const float*)d_in[4];
  const float* V_win      = (const float*)d_in[5];
  const float* FK_win     = (const float*)d_in[6];
  const float* win_score  = (const float*)d_in[7];

  float* out       = (float*)d_out;
  float* out_score = out;                                  // BH*G
  float* out_K     = out_score + (size_t)BH * Gc;          // BH*G*D
  float* out_V     = out_K     + (size_t)BH * Gc * Dd;     // BH*G*D
  float* out_FK    = out_V     + (size_t)BH * Gc * Dd;     // BH*G*F
  float* out_H     = out_FK    + (size_t)BH * Gc * Ff;     // BH*F*D
  float* out_S     = out_H     + (size_t)BH * Ff * Dd;     // BH*F

  int* top_idx = (int*)d_ws;                               // BH*G ints
  int* bot_idx = top_idx + (size_t)BH * Gc;                // BH*C ints

  rank_kernel<<<dim3(Nc / 256, BH), 256, 0, stream>>>(
      heap_score, win_score, out_score, top_idx, bot_idx);

  gather_kernel<<<dim3(Gc, BH), 128, 0, stream>>>(
      K_top, V_top, FK_top, K_win, V_win, FK_win, top_idx,
      out_K, out_V, out_FK);

  hsum_kernel<<<dim3(BH), 256, 0, stream>>>(
      FK_top, FK_win, V_top, V_win, bot_idx, out_H, out_S);
}